// EfficientPnP_17669495455816
// MI455X (gfx1250) — compile-verified
//
#include <hip/hip_runtime.h>
#include <math.h>

#define NPTS 256

typedef float v2f __attribute__((ext_vector_type(2)));
typedef float v8f __attribute__((ext_vector_type(8)));

__device__ __forceinline__ float sgnf(float x) { return (x > 0.f) ? 1.f : ((x < 0.f) ? -1.f : 0.f); }

// ---------------- small dense helpers (per-thread, scratch-resident) ----------------

// Cyclic Jacobi eigensolver for symmetric n x n. A is overwritten (eigenvalues on
// diagonal), V gets eigenvectors as COLUMNS.
__device__ void jacobi_sym(float* A, float* V, int n, int sweeps) {
    for (int i = 0; i < n * n; ++i) V[i] = 0.f;
    for (int i = 0; i < n; ++i) V[i * n + i] = 1.f;
    for (int s = 0; s < sweeps; ++s) {
        for (int p = 0; p < n - 1; ++p) {
            for (int q = p + 1; q < n; ++q) {
                float apq = A[p * n + q];
                if (fabsf(apq) < 1e-12f) continue;
                float app = A[p * n + p], aqq = A[q * n + q];
                float theta = 0.5f * (aqq - app) / apq;
                float t = sgnf(theta) / (fabsf(theta) + sqrtf(theta * theta + 1.f));
                if (theta == 0.f) t = 1.f;
                float c = rsqrtf(t * t + 1.f);
                float sn = t * c;
                for (int k = 0; k < n; ++k) {
                    float akp = A[k * n + p], akq = A[k * n + q];
                    A[k * n + p] = c * akp - sn * akq;
                    A[k * n + q] = sn * akp + c * akq;
                }
                for (int k = 0; k < n; ++k) {
                    float apk = A[p * n + k], aqk = A[q * n + k];
                    A[p * n + k] = c * apk - sn * aqk;
                    A[q * n + k] = sn * apk + c * aqk;
                }
                for (int k = 0; k < n; ++k) {
                    float vkp = V[k * n + p], vkq = V[k * n + q];
                    V[k * n + p] = c * vkp - sn * vkq;
                    V[k * n + q] = sn * vkp + c * vkq;
                }
            }
        }
    }
}

// Gauss-Jordan inverse of 4x4 with partial pivoting.
__device__ void inv4(const float* A, float* out) {
    float M[4][8];
    for (int i = 0; i < 4; ++i)
        for (int j = 0; j < 4; ++j) {
            M[i][j] = A[i * 4 + j];
            M[i][4 + j] = (i == j) ? 1.f : 0.f;
        }
    for (int col = 0; col < 4; ++col) {
        int piv = col;
        float best = fabsf(M[col][col]);
        for (int r = col + 1; r < 4; ++r)
            if (fabsf(M[r][col]) > best) { best = fabsf(M[r][col]); piv = r; }
        if (piv != col)
            for (int j = 0; j < 8; ++j) { float tmp = M[col][j]; M[col][j] = M[piv][j]; M[piv][j] = tmp; }
        float d = M[col][col];
        float inv = 1.f / ((fabsf(d) > 1e-20f) ? d : 1e-20f);
        for (int j = 0; j < 8; ++j) M[col][j] *= inv;
        for (int r = 0; r < 4; ++r) {
            if (r == col) continue;
            float f = M[r][col];
            for (int j = 0; j < 8; ++j) M[r][j] -= f * M[col][j];
        }
    }
    for (int i = 0; i < 4; ++i)
        for (int j = 0; j < 4; ++j) out[i * 4 + j] = M[i][4 + j];
}

// Gauss-Jordan solve A x = b (n x n, n <= 6), A and b clobbered, result in x.
__device__ void gauss_solve(float* A, float* b, float* x, int n) {
    for (int col = 0; col < n; ++col) {
        int piv = col;
        float best = fabsf(A[col * n + col]);
        for (int r = col + 1; r < n; ++r)
            if (fabsf(A[r * n + col]) > best) { best = fabsf(A[r * n + col]); piv = r; }
        if (piv != col) {
            for (int j = 0; j < n; ++j) { float tmp = A[col * n + j]; A[col * n + j] = A[piv * n + j]; A[piv * n + j] = tmp; }
            float tb = b[col]; b[col] = b[piv]; b[piv] = tb;
        }
        float d = A[col * n + col];
        float inv = 1.f / ((fabsf(d) > 1e-20f) ? d : 1e-20f);
        for (int j = 0; j < n; ++j) A[col * n + j] *= inv;
        b[col] *= inv;
        for (int r = 0; r < n; ++r) {
            if (r == col) continue;
            float f = A[r * n + col];
            for (int j = 0; j < n; ++j) A[r * n + j] -= f * A[col * n + j];
            b[r] -= f * b[col];
        }
    }
    for (int i = 0; i < n; ++i) x[i] = b[i];
}

// Procrustes rotation from 3x3 cross-covariance m (row-major): R = U V^T of SVD(m),
// negated wholesale if det < 0 (matches reference).
__device__ void procrustes3(const float* m, float* R) {
    float A[9];
    for (int i = 0; i < 3; ++i)
        for (int j = 0; j < 3; ++j) {
            float s = 0.f;
            for (int k = 0; k < 3; ++k) s += m[k * 3 + i] * m[k * 3 + j];
            A[i * 3 + j] = s;
        }
    float V[9];
    jacobi_sym(A, V, 3, 10);
    float U[9];
    for (int col = 0; col < 3; ++col) {
        float lam = fmaxf(A[col * 3 + col], 0.f);
        float inv = (lam > 1e-20f) ? rsqrtf(lam) : 0.f;
        for (int r = 0; r < 3; ++r) {
            float s = m[r * 3 + 0] * V[0 * 3 + col] + m[r * 3 + 1] * V[1 * 3 + col] + m[r * 3 + 2] * V[2 * 3 + col];
            U[r * 3 + col] = s * inv;
        }
    }
    for (int r = 0; r < 3; ++r)
        for (int c = 0; c < 3; ++c) {
            float s = 0.f;
            for (int k = 0; k < 3; ++k) s += U[r * 3 + k] * V[c * 3 + k];
            R[r * 3 + c] = s;
        }
    float det = R[0] * (R[4] * R[8] - R[5] * R[7]) - R[1] * (R[3] * R[8] - R[5] * R[6]) + R[2] * (R[3] * R[7] - R[4] * R[6]);
    if (det < 0.f)
        for (int i = 0; i < 9; ++i) R[i] = -R[i];
}

__device__ __forceinline__ float block_reduce_sum(float v, float* sRed) {
    int t = threadIdx.x;
    sRed[t] = v;
    __syncthreads();
    for (int s = NPTS / 2; s > 0; s >>= 1) {
        if (t < s) sRed[t] += sRed[t + s];
        __syncthreads();
    }
    float r = sRed[0];
    __syncthreads();
    return r;
}

__constant__ int cSIXP0[6] = {0, 0, 0, 1, 1, 2};
__constant__ int cSIXP1[6] = {1, 2, 3, 2, 3, 3};
__constant__ int cTENP0[10] = {0, 0, 1, 0, 1, 2, 0, 1, 2, 3};
__constant__ int cTENP1[10] = {0, 1, 1, 2, 2, 2, 3, 3, 3, 3};
__constant__ float cMULT[10] = {1.f, 2.f, 1.f, 2.f, 2.f, 1.f, 2.f, 2.f, 2.f, 1.f};

// ---------------- Kernel 1: control points / rho / dw_ctrl / inv(Ch^T) ----------------
__global__ void epnp_prep(const float* __restrict__ obj, float* __restrict__ wCw,
                          float* __restrict__ wCtr, float* __restrict__ wRho,
                          float* __restrict__ wInvA, float* __restrict__ wDwc) {
    __shared__ float sRed[NPTS];
    int b = blockIdx.x, n = threadIdx.x;
    const float* p = obj + (size_t)(b * NPTS + n) * 3;
    float px = p[0], py = p[1], pz = p[2];
    float cx = block_reduce_sum(px, sRed) * (1.f / NPTS);
    float cy = block_reduce_sum(py, sRed) * (1.f / NPTS);
    float cz = block_reduce_sum(pz, sRed) * (1.f / NPTS);
    float dx = px - cx, dy = py - cy, dz = pz - cz;
    float sxx = block_reduce_sum(dx * dx, sRed);
    float sxy = block_reduce_sum(dx * dy, sRed);
    float sxz = block_reduce_sum(dx * dz, sRed);
    float syy = block_reduce_sum(dy * dy, sRed);
    float syz = block_reduce_sum(dy * dz, sRed);
    float szz = block_reduce_sum(dz * dz, sRed);
    if (n == 0) {
        float A[9] = {sxx, sxy, sxz, sxy, syy, syz, sxz, syz, szz};
        float V[9];
        jacobi_sym(A, V, 3, 10);
        float ev[3] = {A[0], A[4], A[8]};
        int ord[3] = {0, 1, 2};
        for (int i = 0; i < 2; ++i)
            for (int j = i + 1; j < 3; ++j)
                if (ev[ord[j]] > ev[ord[i]]) { int t = ord[i]; ord[i] = ord[j]; ord[j] = t; }
        float cw[4][3];
        cw[0][0] = cx; cw[0][1] = cy; cw[0][2] = cz;
        for (int i = 0; i < 3; ++i) {
            int c = ord[i];
            float s = sqrtf(fmaxf(ev[c], 0.f));
            cw[1 + i][0] = cx + s * V[0 * 3 + c];
            cw[1 + i][1] = cy + s * V[1 * 3 + c];
            cw[1 + i][2] = cz + s * V[2 * 3 + c];
        }
        for (int i = 0; i < 4; ++i)
            for (int k = 0; k < 3; ++k) wCw[b * 12 + i * 3 + k] = cw[i][k];
        wCtr[b * 3 + 0] = cx; wCtr[b * 3 + 1] = cy; wCtr[b * 3 + 2] = cz;
        for (int s6 = 0; s6 < 6; ++s6) {
            float d0 = cw[cSIXP0[s6]][0] - cw[cSIXP1[s6]][0];
            float d1 = cw[cSIXP0[s6]][1] - cw[cSIXP1[s6]][1];
            float d2 = cw[cSIXP0[s6]][2] - cw[cSIXP1[s6]][2];
            wRho[b * 6 + s6] = d0 * d0 + d1 * d1 + d2 * d2;
        }
        for (int pp = 0; pp < 4; ++pp)
            for (int q = 0; q < 4; ++q) {
                float d0 = cw[q][0] - cw[pp][0], d1 = cw[q][1] - cw[pp][1], d2 = cw[q][2] - cw[pp][2];
                wDwc[b * 16 + pp * 4 + q] = d0 * d0 + d1 * d1 + d2 * d2;
            }
        float ChT[16], inv[16];
        for (int r = 0; r < 3; ++r)
            for (int j = 0; j < 4; ++j) ChT[r * 4 + j] = cw[j][r];
        for (int j = 0; j < 4; ++j) ChT[3 * 4 + j] = 1.f;
        inv4(ChT, inv);
        for (int i = 0; i < 16; ++i) wInvA[b * 16 + i] = inv[i];
    }
}

// ---------------- Kernel 2: M^T M via V_WMMA_F32_16X16X4_F32 (one wave per batch) ----
// M is (2N x 12), padded to 16 columns; K-loop over 512 rows in chunks of 4. Gram
// product => A fragment (16x4 = M^T chunk) and B fragment (4x16 = M chunk) hold the
// same values under the ISA's mirrored lane striping (lanes 0-15: K=0,1 in the two
// VGPRs; lanes 16-31: K=2,3), so one v2f feeds both operands.
//
// To keep the inner loop branchless, we pre-expand each point's (row_u, row_v) pair
// for all 16 padded columns into LDS, interleaved as [point][col][2] so each lane
// fetches its (u,v) pair with a single 8-byte-aligned ds_load_b64. Lane halves read
// adjacent points => disjoint LDS bank ranges, no conflicts.
__global__ void epnp_mtm(const float* __restrict__ obj, const float* __restrict__ img,
                         const float* __restrict__ intr, const float* __restrict__ wInvA,
                         float* __restrict__ wMtm) {
    __shared__ float sM[NPTS * 32];  // [point][col 0..15][u,v]  = 32 KB
    int b = blockIdx.x, lane = threadIdx.x;
    float fu = intr[b * 9 + 0], cxi = intr[b * 9 + 2];
    float fv = intr[b * 9 + 4], cyi = intr[b * 9 + 5];
    float iA[16];
#pragma unroll
    for (int i = 0; i < 16; ++i) iA[i] = wInvA[b * 16 + i];

    // Stage: expand M rows for this batch into LDS (uniform trip count, EXEC stays full).
    for (int p = lane; p < NPTS; p += 32) {
        const float* o = obj + (size_t)(b * NPTS + p) * 3;
        float h0 = o[0], h1 = o[1], h2 = o[2];
        float a[4];
#pragma unroll
        for (int j = 0; j < 4; ++j)
            a[j] = iA[j * 4 + 0] * h0 + iA[j * 4 + 1] * h1 + iA[j * 4 + 2] * h2 + iA[j * 4 + 3];
        const float* im = img + (size_t)(b * NPTS + p) * 2;
        float du = cxi - im[0];
        float dv = cyi - im[1];
        float* row = sM + p * 32;
#pragma unroll
        for (int c = 0; c < 16; ++c) {  // j, comp become compile-time constants
            int j = c / 3, comp = c - 3 * j;
            float uu = 0.f, vv = 0.f;
            if (c < 12) {
                uu = (comp == 0) ? a[j] * fu : ((comp == 2) ? a[j] * du : 0.f);
                vv = (comp == 1) ? a[j] * fv : ((comp == 2) ? a[j] * dv : 0.f);
            }
            row[c * 2 + 0] = uu;
            row[c * 2 + 1] = vv;
        }
    }
    __syncthreads();

    int half = lane >> 4;   // 0: K=0,1 (point 2*it)   1: K=2,3 (point 2*it+1)
    int idx = lane & 15;    // M^T row index == M column index
    const float* lanePtr = sM + half * 32 + idx * 2;
    v8f acc = {0.f, 0.f, 0.f, 0.f, 0.f, 0.f, 0.f, 0.f};
    for (int it = 0; it < NPTS / 2; ++it) {
        v2f f = *(const v2f*)(lanePtr + it * 64);  // ds_load_b64: (row_u, row_v)
        acc = __builtin_amdgcn_wmma_f32_16x16x4_f32(false, f, false, f, (short)0, acc,
                                                    false, false);
    }
    // C/D layout: lanes 0-15 -> M = vgpr, lanes 16-31 -> M = 8 + vgpr; N = lane & 15.
#pragma unroll
    for (int r = 0; r < 8; ++r) {
        int Mr = r + (half ? 8 : 0);
        if (Mr < 12 && idx < 12) wMtm[(size_t)b * 144 + Mr * 12 + idx] = acc[r];
    }
}

// ---------------- Kernel 3: 12x12 eigensolve -> km (12x4) --------------------------
__global__ void epnp_eig(const float* __restrict__ wMtm, float* __restrict__ wKm, int B) {
    int t = blockIdx.x * blockDim.x + threadIdx.x;
    if (t >= B) return;
    float A[144], V[144];
    for (int i = 0; i < 144; ++i) A[i] = wMtm[(size_t)t * 144 + i];
    jacobi_sym(A, V, 12, 10);
    float ev[12];
    int ord[12];
    for (int i = 0; i < 12; ++i) { ev[i] = A[i * 12 + i]; ord[i] = i; }
    for (int i = 0; i < 11; ++i)          // ascending
        for (int j = i + 1; j < 12; ++j)
            if (ev[ord[j]] < ev[ord[i]]) { int tmp = ord[i]; ord[i] = ord[j]; ord[j] = tmp; }
    // km[:, j] = eigvec with (4-j)-th smallest eigenvalue  (v[:, :4][:, ::-1])
    for (int r = 0; r < 12; ++r)
        for (int j = 0; j < 4; ++j)
            wKm[(size_t)t * 48 + r * 4 + j] = V[r * 12 + ord[3 - j]];
}

// ---------------- Kernel 4: L matrix + betas for dim 1/2/3 -------------------------
__global__ void epnp_betas(const float* __restrict__ wKm, const float* __restrict__ wRho,
                           float* __restrict__ wBet, int B) {
    int t = blockIdx.x * blockDim.x + threadIdx.x;
    if (t >= B) return;
    const float* km = wKm + (size_t)t * 48;
    float dv[4][6][3];
    for (int a = 0; a < 4; ++a)
        for (int s = 0; s < 6; ++s)
            for (int x = 0; x < 3; ++x)
                dv[a][s][x] = km[(cSIXP1[s] * 3 + x) * 4 + a] - km[(cSIXP0[s] * 3 + x) * 4 + a];
    float L[6][10];
    for (int s = 0; s < 6; ++s)
        for (int tt = 0; tt < 10; ++tt) {
            int a0 = cTENP0[tt], a1 = cTENP1[tt];
            float d = dv[a0][s][0] * dv[a1][s][0] + dv[a0][s][1] * dv[a1][s][1] + dv[a0][s][2] * dv[a1][s][2];
            L[s][tt] = d * cMULT[tt];
        }
    float rho[6];
    for (int s = 0; s < 6; ++s) rho[s] = wRho[t * 6 + s];
    float* bet = wBet + (size_t)t * 12;
    // dim 1
    bet[0] = 0.f; bet[1] = 0.f; bet[2] = 0.f; bet[3] = 1.f;
    // dim 2 : pinv of L[:, (5,8,9)] via normal equations
    {
        const int cols[3] = {5, 8, 9};
        float G[9], h[3], bt[3];
        for (int i = 0; i < 3; ++i) {
            h[i] = 0.f;
            for (int j = 0; j < 3; ++j) {
                float s = 0.f;
                for (int r = 0; r < 6; ++r) s += L[r][cols[i]] * L[r][cols[j]];
                G[i * 3 + j] = s;
            }
            for (int r = 0; r < 6; ++r) h[i] += L[r][cols[i]] * rho[r];
        }
        gauss_solve(G, h, bt, 3);
        float b1 = sqrtf(fabsf(bt[0]));
        float b2 = sqrtf(fabsf(bt[2])) * sgnf(bt[1]) * sgnf(bt[0]);
        bet[4] = 0.f; bet[5] = 0.f; bet[6] = b1; bet[7] = b2;
    }
    // dim 3 : solve L[:, (2,4,7,5,8,9)] x = rho
    {
        const int cols[6] = {2, 4, 7, 5, 8, 9};
        float M6[36], r6[6], bt[6];
        for (int r = 0; r < 6; ++r) {
            r6[r] = rho[r];
            for (int c = 0; c < 6; ++c) M6[r * 6 + c] = L[r][cols[c]];
        }
        gauss_solve(M6, r6, bt, 6);
        float b1 = sqrtf(fabsf(bt[0]));
        float b2 = sqrtf(fabsf(bt[3])) * sgnf(bt[1]) * sgnf(bt[0]);
        float b3 = sqrtf(fabsf(bt[5])) * sgnf(bt[2]) * sgnf(bt[0]);
        bet[8] = 0.f; bet[9] = b1; bet[10] = b2; bet[11] = b3;
    }
}

// ---------------- Kernel 5: candidates + Gauss-Newton + outputs --------------------
__global__ void epnp_cand(const float* __restrict__ obj, const float* __restrict__ img,
                          const float* __restrict__ intr, const float* __restrict__ wKm,
                          const float* __restrict__ wCtr, const float* __restrict__ wInvA,
                          const float* __restrict__ wDwc, const float* __restrict__ wBet,
                          float* __restrict__ out, int B) {
    __shared__ float sRed[NPTS];
    __shared__ float sObj[NPTS][3];
    __shared__ float sImg[NPTS][2];
    __shared__ float sAl[NPTS][4];
    __shared__ float sDw[NPTS];
    __shared__ float sOcB[NPTS][3];
    __shared__ float sKm[48], sWm[3], sDwc[16], sInvA[16], sBc[12];
    __shared__ float sCc[12], sR[9], sT[3], sBeta[4], sBetaGN[4];
    __shared__ float sBErr, sBR[9], sBT[3], sBCc[12], sBBeta[4], sBScale;
    __shared__ float sSc, sScS;
    __shared__ int sFlag;

    int b = blockIdx.x, n = threadIdx.x;
    {
        const float* o = obj + (size_t)(b * NPTS + n) * 3;
        sObj[n][0] = o[0]; sObj[n][1] = o[1]; sObj[n][2] = o[2];
        const float* im = img + (size_t)(b * NPTS + n) * 2;
        sImg[n][0] = im[0]; sImg[n][1] = im[1];
    }
    if (n < 48) sKm[n] = wKm[(size_t)b * 48 + n];
    if (n < 16) { sDwc[n] = wDwc[b * 16 + n]; sInvA[n] = wInvA[b * 16 + n]; }
    if (n < 12) sBc[n] = wBet[(size_t)b * 12 + n];
    if (n < 3) sWm[n] = wCtr[b * 3 + n];
    if (n == 0) sBErr = 3.4e38f;
    __syncthreads();
    {
        float h0 = sObj[n][0], h1 = sObj[n][1], h2 = sObj[n][2];
        for (int j = 0; j < 4; ++j)
            sAl[n][j] = sInvA[j * 4 + 0] * h0 + sInvA[j * 4 + 1] * h1 + sInvA[j * 4 + 2] * h2 + sInvA[j * 4 + 3];
        float dx = h0 - sWm[0], dy = h1 - sWm[1], dz = h2 - sWm[2];
        sDw[n] = sqrtf(dx * dx + dy * dy + dz * dz);
    }
    float fu = intr[b * 9 + 0], cxi = intr[b * 9 + 2];
    float fv = intr[b * 9 + 4], cyi = intr[b * 9 + 5];
    __syncthreads();

    for (int trial = 0; trial < 13; ++trial) {
        if (n == 0) {
            if (trial < 3) {
                for (int m = 0; m < 4; ++m) sBeta[m] = sBc[trial * 4 + m];
            } else {
                // Gauss-Newton step: r = mean |dw - dc|, closed-form gradient.
                float cc[12];
                for (int i = 0; i < 12; ++i) {
                    float s = 0.f;
                    for (int m = 0; m < 4; ++m) s += sKm[i * 4 + m] * sBetaGN[m];
                    cc[i] = s;
                }
                float r = 0.f, g[4] = {0.f, 0.f, 0.f, 0.f};
                for (int p = 0; p < 4; ++p)
                    for (int q = 0; q < 4; ++q) {
                        float d0 = cc[q * 3 + 0] - cc[p * 3 + 0];
                        float d1 = cc[q * 3 + 1] - cc[p * 3 + 1];
                        float d2 = cc[q * 3 + 2] - cc[p * 3 + 2];
                        float dc = d0 * d0 + d1 * d1 + d2 * d2;
                        float diff = sDwc[p * 4 + q] - dc;
                        r += fabsf(diff);
                        float s = -sgnf(diff) * 2.f;
                        for (int m = 0; m < 4; ++m) {
                            float e0 = sKm[(q * 3 + 0) * 4 + m] - sKm[(p * 3 + 0) * 4 + m];
                            float e1 = sKm[(q * 3 + 1) * 4 + m] - sKm[(p * 3 + 1) * 4 + m];
                            float e2 = sKm[(q * 3 + 2) * 4 + m] - sKm[(p * 3 + 2) * 4 + m];
                            g[m] += s * (d0 * e0 + d1 * e1 + d2 * e2);
                        }
                    }
                r *= (1.f / 16.f);
                float gg = 1e-12f;
                for (int m = 0; m < 4; ++m) { g[m] *= (1.f / 16.f); gg += g[m] * g[m]; }
                float step = r / gg;
                for (int m = 0; m < 4; ++m) { sBetaGN[m] -= g[m] * step; sBeta[m] = sBetaGN[m]; }
            }
            for (int i = 0; i < 12; ++i) {
                float s = 0.f;
                for (int m = 0; m < 4; ++m) s += sKm[i * 4 + m] * sBeta[m];
                sCc[i] = s;
            }
        }
        __syncthreads();

        float a0 = sAl[n][0], a1 = sAl[n][1], a2 = sAl[n][2], a3 = sAl[n][3];
        float ocx = a0 * sCc[0] + a1 * sCc[3] + a2 * sCc[6] + a3 * sCc[9];
        float ocy = a0 * sCc[1] + a1 * sCc[4] + a2 * sCc[7] + a3 * sCc[10];
        float ocz = a0 * sCc[2] + a1 * sCc[5] + a2 * sCc[8] + a3 * sCc[11];
        float omx = block_reduce_sum(ocx, sRed) * (1.f / NPTS);
        float omy = block_reduce_sum(ocy, sRed) * (1.f / NPTS);
        float omz = block_reduce_sum(ocz, sRed) * (1.f / NPTS);
        float dcx = ocx - omx, dcy = ocy - omy, dcz = ocz - omz;
        float dc = sqrtf(dcx * dcx + dcy * dcy + dcz * dcz);
        float num = block_reduce_sum(dc * sDw[n], sRed);
        float den = block_reduce_sum(dc * dc, sRed);
        float sc = num / fmaxf(den, 1e-12f);
        float negc = block_reduce_sum((ocz * sc < 0.f) ? 1.f : 0.f, sRed);
        float sw = (negc > 0.f) ? -1.f : 1.f;
        float ss = sc * sw;
        float opx = ocx * ss, opy = ocy * ss, opz = ocz * ss;
        float cmx = omx * ss, cmy = omy * ss, cmz = omz * ss;
        float wx = sObj[n][0] - sWm[0], wy = sObj[n][1] - sWm[1], wz = sObj[n][2] - sWm[2];
        float px_ = opx - cmx, py_ = opy - cmy, pz_ = opz - cmz;
        float m00 = block_reduce_sum(px_ * wx, sRed);
        float m01 = block_reduce_sum(px_ * wy, sRed);
        float m02 = block_reduce_sum(px_ * wz, sRed);
        float m10 = block_reduce_sum(py_ * wx, sRed);
        float m11 = block_reduce_sum(py_ * wy, sRed);
        float m12 = block_reduce_sum(py_ * wz, sRed);
        float m20 = block_reduce_sum(pz_ * wx, sRed);
        float m21 = block_reduce_sum(pz_ * wy, sRed);
        float m22 = block_reduce_sum(pz_ * wz, sRed);
        if (n == 0) {
            float m[9] = {m00, m01, m02, m10, m11, m12, m20, m21, m22};
            float R[9];
            procrustes3(m, R);
            for (int i = 0; i < 9; ++i) sR[i] = R[i];
            sT[0] = cmx - (R[0] * sWm[0] + R[1] * sWm[1] + R[2] * sWm[2]);
            sT[1] = cmy - (R[3] * sWm[0] + R[4] * sWm[1] + R[5] * sWm[2]);
            sT[2] = cmz - (R[6] * sWm[0] + R[7] * sWm[1] + R[8] * sWm[2]);
            sSc = sc;
            sScS = ss;
        }
        __syncthreads();
        float ox = sObj[n][0], oy = sObj[n][1], oz = sObj[n][2];
        float camx = sR[0] * ox + sR[1] * oy + sR[2] * oz + sT[0];
        float camy = sR[3] * ox + sR[4] * oy + sR[5] * oz + sT[1];
        float camz = sR[6] * ox + sR[7] * oy + sR[8] * oz + sT[2];
        float invz = 1.f / camz;
        float u = (fu * camx + cxi * camz) * invz;
        float v = (fv * camy + cyi * camz) * invz;
        float du = u - sImg[n][0], dvv = v - sImg[n][1];
        float err = block_reduce_sum(sqrtf(du * du + dvv * dvv), sRed) * (1.f / NPTS);
        if (n == 0) {
            sFlag = 0;
            if (err < sBErr) {
                sBErr = err;
                for (int i = 0; i < 9; ++i) sBR[i] = sR[i];
                for (int i = 0; i < 3; ++i) sBT[i] = sT[i];
                for (int i = 0; i < 12; ++i) sBCc[i] = sCc[i] * sSc;  // cc scaled, not sign-flipped
                for (int m = 0; m < 4; ++m) sBBeta[m] = sBeta[m];
                sBScale = sScS;
                sFlag = 1;
            }
            if (trial == 2)
                for (int m = 0; m < 4; ++m) sBetaGN[m] = sBBeta[m] * sBScale;
        }
        __syncthreads();
        if (sFlag) { sOcB[n][0] = opx; sOcB[n][1] = opy; sOcB[n][2] = opz; }
        __syncthreads();
    }

    // Outputs: error | R | t | contPts_c | objPts_c | beta | scale (flat, in order)
    float* oErr = out;
    float* oR = out + B;
    float* oT = oR + (size_t)B * 9;
    float* oCc = oT + (size_t)B * 3;
    float* oOc = oCc + (size_t)B * 12;
    float* oBeta = oOc + (size_t)B * NPTS * 3;
    float* oSc = oBeta + (size_t)B * 4;
    if (n == 0) {
        oErr[b] = sBErr;
        for (int i = 0; i < 9; ++i) oR[b * 9 + i] = sBR[i];
        for (int i = 0; i < 3; ++i) oT[b * 3 + i] = sBT[i];
        for (int i = 0; i < 12; ++i) oCc[b * 12 + i] = sBCc[i];
        for (int m = 0; m < 4; ++m) oBeta[b * 4 + m] = sBBeta[m];
        oSc[b] = sBScale;
    }
    for (int k = 0; k < 3; ++k) oOc[((size_t)b * NPTS + n) * 3 + k] = sOcB[n][k];
}

// ---------------- host glue --------------------------------------------------------
extern "C" void kernel_launch(void* const* d_in, const int* in_sizes, int n_in,
                              void* d_out, int out_size, void* d_ws, size_t ws_size,
                              hipStream_t stream) {
    const float* obj = (const float*)d_in[0];
    const float* img = (const float*)d_in[1];
    const float* intr = (const float*)d_in[2];
    float* out = (float*)d_out;
    int B = in_sizes[0] / (NPTS * 3);

    // Workspace layout (floats): 257 per batch (~4.2 MB at B=4096).
    float* w = (float*)d_ws;
    float* wCw = w;                            // B*12
    float* wCtr = wCw + (size_t)B * 12;        // B*3
    float* wRho = wCtr + (size_t)B * 3;        // B*6
    float* wInvA = wRho + (size_t)B * 6;       // B*16
    float* wDwc = wInvA + (size_t)B * 16;      // B*16
    float* wMtm = wDwc + (size_t)B * 16;       // B*144
    float* wKm = wMtm + (size_t)B * 144;       // B*48
    float* wBet = wKm + (size_t)B * 48;        // B*12

    epnp_prep<<<B, NPTS, 0, stream>>>(obj, wCw, wCtr, wRho, wInvA, wDwc);
    epnp_mtm<<<B, 32, 0, stream>>>(obj, img, intr, wInvA, wMtm);
    int blk = (B + 255) / 256;
    epnp_eig<<<blk, 256, 0, stream>>>(wMtm, wKm, B);
    epnp_betas<<<blk, 256, 0, stream>>>(wKm, wRho, wBet, B);
    epnp_cand<<<B, NPTS, 0, stream>>>(obj, img, intr, wKm, wCtr, wInvA, wDwc, wBet, out, B);
}